// DirSageConv_28054726378292
// MI455X (gfx1250) — compile-verified
//
#include <hip/hip_runtime.h>
#include <cmath>

#define N_NODES 100000
#define N_EDGES 1000000
#define F_IN    64
#define F_OUT   128
#define F_HID   512

typedef __attribute__((ext_vector_type(2))) float v2f;
typedef __attribute__((ext_vector_type(8))) float v8f;

__device__ __forceinline__ v8f wmma_f32(v2f a, v2f b, v8f c) {
    // D = A(16x4 f32) x B(4x16 f32) + C(16x16 f32)
    return __builtin_amdgcn_wmma_f32_16x16x4_f32(
        /*neg_a=*/false, a, /*neg_b=*/false, b,
        /*c_mod=*/(short)0, c, /*reuse_a=*/false, /*reuse_b=*/false);
}

__device__ __forceinline__ float elu1(float x) {
    return x > 0.0f ? x : expm1f(x);
}

// ---------------------------------------------------------------------------
// 1) zero the aggregation workspace
// ---------------------------------------------------------------------------
__global__ void dsc_zero_kernel(float* ws, int n) {
    int i = blockIdx.x * blockDim.x + threadIdx.x;
    if (i < n) ws[i] = 0.0f;
}

// ---------------------------------------------------------------------------
// 2) edge scatter: 16 threads per edge, float4 per thread, both directions
// ---------------------------------------------------------------------------
__global__ void dsc_scatter_kernel(const float* __restrict__ x,
                                   const long long* __restrict__ ei,
                                   float* __restrict__ agg_in,
                                   float* __restrict__ cnt_in,
                                   float* __restrict__ agg_out,
                                   float* __restrict__ cnt_out) {
    long long t = (long long)blockIdx.x * blockDim.x + threadIdx.x;
    int e = (int)(t >> 4);
    int g = (int)(t & 15);
    if (e >= N_EDGES) return;
    int src = (int)ei[e];
    int dst = (int)ei[N_EDGES + e];

    // x[src] aggregated at dst  (x_in branch)
    float4 vs = ((const float4*)(x + (size_t)src * F_IN))[g];
    float* ai = agg_in + (size_t)dst * F_IN + g * 4;
    atomicAdd(ai + 0, vs.x);
    atomicAdd(ai + 1, vs.y);
    atomicAdd(ai + 2, vs.z);
    atomicAdd(ai + 3, vs.w);

    // x[dst] aggregated at src  (x_out branch)
    float4 vd = ((const float4*)(x + (size_t)dst * F_IN))[g];
    float* ao = agg_out + (size_t)src * F_IN + g * 4;
    atomicAdd(ao + 0, vd.x);
    atomicAdd(ao + 1, vd.y);
    atomicAdd(ao + 2, vd.z);
    atomicAdd(ao + 3, vd.w);

    if (g == 0) {
        atomicAdd(cnt_in + dst, 1.0f);
        atomicAdd(cnt_out + src, 1.0f);
    }
}

// ---------------------------------------------------------------------------
// 3) elu((agg/deg) @ W + bias) : one wave -> one 16x16 output tile
//    grid = (N_NODES/16, F_OUT/16), block = 32
// ---------------------------------------------------------------------------
__global__ void __launch_bounds__(32)
dsc_sage_gemm_kernel(const float* __restrict__ agg,
                     const float* __restrict__ cnt,
                     const float* __restrict__ W,   // F_IN x F_OUT
                     const float* __restrict__ bias,
                     float* __restrict__ out) {
    const int lane = threadIdx.x;
    const int row0 = blockIdx.x * 16;
    const int nt   = blockIdx.y;

    const int ln   = lane & 15;           // A-row for this lane / N-col offset
    const int koff = (lane >> 4) * 2;     // K sub-offset per half-wave
    const int n    = nt * 16 + ln;

    const float rc = 1.0f / fmaxf(cnt[row0 + ln], 1.0f);
    const float* arow = agg + (size_t)(row0 + ln) * F_IN;

    v8f c0 = {}; v8f c1 = {};             // two independent K chains for ILP
#pragma unroll
    for (int kb = 0; kb < F_IN / 2; kb += 4) {
        v2f a0, b0, a1, b1v;
        a0.x = arow[kb + koff] * rc;
        a0.y = arow[kb + koff + 1] * rc;
        b0.x = W[(kb + koff) * F_OUT + n];
        b0.y = W[(kb + koff + 1) * F_OUT + n];
        c0 = wmma_f32(a0, b0, c0);

        a1.x = arow[F_IN / 2 + kb + koff] * rc;
        a1.y = arow[F_IN / 2 + kb + koff + 1] * rc;
        b1v.x = W[(F_IN / 2 + kb + koff) * F_OUT + n];
        b1v.y = W[(F_IN / 2 + kb + koff + 1) * F_OUT + n];
        c1 = wmma_f32(a1, b1v, c1);
    }

    const float bn = bias[n];
#pragma unroll
    for (int r = 0; r < 8; ++r) {
        int m = r + ((lane >> 4) << 3);
        float v = c0[r] + c1[r] + bn;
        out[(size_t)(row0 + m) * F_OUT + n] = elu1(v);
    }
}

// ---------------------------------------------------------------------------
// 4) fused MLP: elu(elu(x@W1+b1)@W2+b2), 16 rows per block, 8 waves
// ---------------------------------------------------------------------------
#define XSTR 68    // 16x64 tile, padded (4*m mod 64 distinct banks)
#define HSTR 516   // 16x512 hidden tile, padded

__global__ void __launch_bounds__(256)
dsc_mlp_kernel(const float* __restrict__ x,
               const float* __restrict__ W1, const float* __restrict__ b1,
               const float* __restrict__ W2, const float* __restrict__ b2,
               float* __restrict__ out) {
    __shared__ float sX[16 * XSTR];
    __shared__ float sH[16 * HSTR];

    const int lane = threadIdx.x & 31;
    const int wv   = threadIdx.x >> 5;       // 0..7
    const int row0 = blockIdx.x * 16;

    // stage x tile into LDS
    for (int i = threadIdx.x; i < 16 * F_IN; i += 256) {
        int m = i >> 6, k = i & 63;
        sX[m * XSTR + k] = x[(size_t)(row0 + m) * F_IN + k];
    }
    __syncthreads();

    const int ln   = lane & 15;
    const int koff = (lane >> 4) * 2;

    // ---- stage 1: H = elu(x @ W1 + b1), each wave does 4 of 32 N-tiles ----
#pragma unroll
    for (int j = 0; j < 4; ++j) {
        const int nt = wv * 4 + j;
        const int n  = nt * 16 + ln;
        v8f c0 = {}; v8f c1 = {};
#pragma unroll
        for (int kb = 0; kb < F_IN / 2; kb += 4) {
            v2f a0, bb0, a1, bb1;
            a0.x = sX[ln * XSTR + kb + koff];
            a0.y = sX[ln * XSTR + kb + koff + 1];
            bb0.x = W1[(kb + koff) * F_HID + n];
            bb0.y = W1[(kb + koff + 1) * F_HID + n];
            c0 = wmma_f32(a0, bb0, c0);

            a1.x = sX[ln * XSTR + F_IN / 2 + kb + koff];
            a1.y = sX[ln * XSTR + F_IN / 2 + kb + koff + 1];
            bb1.x = W1[(F_IN / 2 + kb + koff) * F_HID + n];
            bb1.y = W1[(F_IN / 2 + kb + koff + 1) * F_HID + n];
            c1 = wmma_f32(a1, bb1, c1);
        }
        const float bn = b1[n];
#pragma unroll
        for (int r = 0; r < 8; ++r) {
            int m = r + ((lane >> 4) << 3);
            sH[m * HSTR + n] = elu1(c0[r] + c1[r] + bn);
        }
    }
    __syncthreads();

    // ---- stage 2: out = elu(H @ W2 + b2), wave wv -> N-tile wv (of 8) ----
    const int n = wv * 16 + ln;
    v8f c0 = {}; v8f c1 = {};
#pragma unroll 8
    for (int kb = 0; kb < F_HID / 2; kb += 4) {
        v2f a0, bb0, a1, bb1;
        a0.x = sH[ln * HSTR + kb + koff];
        a0.y = sH[ln * HSTR + kb + koff + 1];
        bb0.x = W2[(kb + koff) * F_OUT + n];
        bb0.y = W2[(kb + koff + 1) * F_OUT + n];
        c0 = wmma_f32(a0, bb0, c0);

        a1.x = sH[ln * HSTR + F_HID / 2 + kb + koff];
        a1.y = sH[ln * HSTR + F_HID / 2 + kb + koff + 1];
        bb1.x = W2[(F_HID / 2 + kb + koff) * F_OUT + n];
        bb1.y = W2[(F_HID / 2 + kb + koff + 1) * F_OUT + n];
        c1 = wmma_f32(a1, bb1, c1);
    }
    const float bn = b2[n];
#pragma unroll
    for (int r = 0; r < 8; ++r) {
        int m = r + ((lane >> 4) << 3);
        out[(size_t)(row0 + m) * F_OUT + n] = elu1(c0[r] + c1[r] + bn);
    }
}

// ---------------------------------------------------------------------------
extern "C" void kernel_launch(void* const* d_in, const int* in_sizes, int n_in,
                              void* d_out, int out_size, void* d_ws, size_t ws_size,
                              hipStream_t stream) {
    const float*     x     = (const float*)d_in[0];
    const long long* ei    = (const long long*)d_in[1];
    const float*     W_in  = (const float*)d_in[2];
    const float*     b_in  = (const float*)d_in[3];
    const float*     W_out = (const float*)d_in[4];
    const float*     b_out = (const float*)d_in[5];
    const float*     W1    = (const float*)d_in[6];
    const float*     b1    = (const float*)d_in[7];
    const float*     W2    = (const float*)d_in[8];
    const float*     b2    = (const float*)d_in[9];

    float* out = (float*)d_out;
    float* ws  = (float*)d_ws;

    float* agg_in  = ws;                                    // N x 64
    float* agg_out = ws + (size_t)N_NODES * F_IN;           // N x 64
    float* cnt_in  = ws + (size_t)2 * N_NODES * F_IN;       // N
    float* cnt_out = cnt_in + N_NODES;                      // N

    const int nzero = 2 * N_NODES * F_IN + 2 * N_NODES;
    dsc_zero_kernel<<<(nzero + 255) / 256, 256, 0, stream>>>(ws, nzero);

    const long long nthreads = (long long)N_EDGES * 16;
    dsc_scatter_kernel<<<(int)((nthreads + 255) / 256), 256, 0, stream>>>(
        x, ei, agg_in, cnt_in, agg_out, cnt_out);

    dim3 gdim(N_NODES / 16, F_OUT / 16);
    // x_in  -> out[0 : N*128)
    dsc_sage_gemm_kernel<<<gdim, 32, 0, stream>>>(agg_in, cnt_in, W_in, b_in, out);
    // x_out -> out[N*128 : 2*N*128)
    dsc_sage_gemm_kernel<<<gdim, 32, 0, stream>>>(agg_out, cnt_out, W_out, b_out,
                                                  out + (size_t)N_NODES * F_OUT);
    // x_self -> out[2*N*128 : 3*N*128)
    dsc_mlp_kernel<<<N_NODES / 16, 256, 0, stream>>>(x, W1, b1, W2, b2,
                                                     out + (size_t)2 * N_NODES * F_OUT);
}